// CLSNet_72224170049964
// MI455X (gfx1250) — compile-verified
//
#include <hip/hip_runtime.h>
#include <math.h>

// ---------------- problem constants (from reference) ----------------
#define B_   16
#define C_   128
#define H_   128
#define W_   128
#define HW_  (H_ * W_)      // 16384
#define K_   6
#define P_   8
#define KP_  (K_ * P_)      // 48
#define NPATCH 256          // B * 4 * 4
#define NPP  1024           // pixels per patch (32x32)
#define EPS_ 1e-6f

typedef __attribute__((ext_vector_type(2))) float v2f;
typedef __attribute__((ext_vector_type(8))) float v8f;

__device__ __forceinline__ float gelu_exact(float x) {
    return 0.5f * x * (1.0f + erff(x * 0.70710678118654752f));
}

// =====================================================================
// K1: per-patch  pred -> softmax(n) -> ctx = probs * feats^T -> LN + GELU
// one workgroup per patch (256 blocks x 256 threads)
// writes center[k][bp][c]  (6 x 256 x 128)
// =====================================================================
__global__ void k1_patch_ctx(const float* __restrict__ feats,
                             const float* __restrict__ coarse_w,
                             const float* __restrict__ coarse_b,
                             const float* __restrict__ ln1_g,
                             const float* __restrict__ ln1_b,
                             float* __restrict__ center) {
    __shared__ float s_pred[K_ * NPP];   // 24 KB  (pred, then exp(pred-max))
    __shared__ float s_buf[8448];        // 33 KB  (redm | fch 128x66 | ctx+stats)
    __shared__ float s_w[784];           // coarse_w(768) + bias(6) + rowinv(6)

    const int t  = threadIdx.x;
    const int bp = blockIdx.x;
    const int b  = bp >> 4;
    const int py = (bp >> 2) & 3;
    const int px = bp & 3;
    const float* fbase = feats + (size_t)b * C_ * HW_ + (py * 32) * W_ + px * 32;

    // stage weights
    for (int i = t; i < K_ * C_; i += 256) s_w[i] = coarse_w[i];
    if (t < K_) s_w[768 + t] = coarse_b[t];
    __syncthreads();

    // ---- phase 1: pred[k][n] for 4 pixels per thread ----
    float acc[4][K_];
    int off[4];
    #pragma unroll
    for (int i = 0; i < 4; ++i) {
        int n = t + i * 256;
        off[i] = (n >> 5) * W_ + (n & 31);
        #pragma unroll
        for (int k = 0; k < K_; ++k) acc[i][k] = s_w[768 + k];
    }
    for (int c = 0; c < C_; ++c) {
        const float* fc = fbase + (size_t)c * HW_;
        float f0 = fc[off[0]], f1 = fc[off[1]], f2 = fc[off[2]], f3 = fc[off[3]];
        #pragma unroll
        for (int k = 0; k < K_; ++k) {
            float w = s_w[k * C_ + c];
            acc[0][k] = fmaf(f0, w, acc[0][k]);
            acc[1][k] = fmaf(f1, w, acc[1][k]);
            acc[2][k] = fmaf(f2, w, acc[2][k]);
            acc[3][k] = fmaf(f3, w, acc[3][k]);
        }
    }
    #pragma unroll
    for (int i = 0; i < 4; ++i)
        #pragma unroll
        for (int k = 0; k < K_; ++k) s_pred[k * NPP + t + i * 256] = acc[i][k];
    __syncthreads();

    // ---- phase 2: softmax over n (max, exp, sum); keep unnormalized e ----
    float pm[K_];
    #pragma unroll
    for (int k = 0; k < K_; ++k) pm[k] = -3.4e38f;
    #pragma unroll
    for (int i = 0; i < 4; ++i) {
        int n = t + i * 256;
        #pragma unroll
        for (int k = 0; k < K_; ++k) pm[k] = fmaxf(pm[k], s_pred[k * NPP + n]);
    }
    #pragma unroll
    for (int k = 0; k < K_; ++k) s_buf[k * 256 + t] = pm[k];
    __syncthreads();
    for (int s = 128; s > 0; s >>= 1) {
        if (t < s)
            #pragma unroll
            for (int k = 0; k < K_; ++k)
                s_buf[k * 256 + t] = fmaxf(s_buf[k * 256 + t], s_buf[k * 256 + t + s]);
        __syncthreads();
    }
    float rmax[K_];
    #pragma unroll
    for (int k = 0; k < K_; ++k) rmax[k] = s_buf[k * 256];
    __syncthreads();

    float ps[K_] = {0, 0, 0, 0, 0, 0};
    #pragma unroll
    for (int i = 0; i < 4; ++i) {
        int n = t + i * 256;
        #pragma unroll
        for (int k = 0; k < K_; ++k) {
            float e = expf(s_pred[k * NPP + n] - rmax[k]);
            s_pred[k * NPP + n] = e;
            ps[k] += e;
        }
    }
    #pragma unroll
    for (int k = 0; k < K_; ++k) s_buf[k * 256 + t] = ps[k];
    __syncthreads();
    for (int s = 128; s > 0; s >>= 1) {
        if (t < s)
            #pragma unroll
            for (int k = 0; k < K_; ++k)
                s_buf[k * 256 + t] += s_buf[k * 256 + t + s];
        __syncthreads();
    }
    if (t < K_) s_w[778 + t] = 1.0f / s_buf[t * 256];
    __syncthreads();

    // ---- phase 3: ctx[k][c] = sum_n probs * feats ; 3 (k,c) per thread ----
    const int c  = t & 127;
    const int kb = t >> 7;                 // 0/1
    const int kk0 = kb, kk1 = kb + 2, kk2 = kb + 4;
    float a0 = 0.f, a1 = 0.f, a2 = 0.f;
    for (int ci = 0; ci < 16; ++ci) {      // chunks of 64 pixels
        __syncthreads();                   // previous chunk fully consumed
        const int n0 = ci * 64;
        for (int i = t; i < C_ * 64; i += 256) {
            int cc = i >> 6, nn = i & 63, n = n0 + nn;
            s_buf[cc * 66 + nn] = fbase[(size_t)cc * HW_ + (n >> 5) * W_ + (n & 31)];
        }
        __syncthreads();
        for (int nn = 0; nn < 64; ++nn) {
            float f = s_buf[c * 66 + nn];
            int n = n0 + nn;
            a0 = fmaf(s_pred[kk0 * NPP + n], f, a0);
            a1 = fmaf(s_pred[kk1 * NPP + n], f, a1);
            a2 = fmaf(s_pred[kk2 * NPP + n], f, a2);
        }
    }
    __syncthreads();

    // ---- phase 4: LN over C (per k) + GELU -> center ----
    float ctx0 = a0 * s_w[778 + kk0];
    float ctx1 = a1 * s_w[778 + kk1];
    float ctx2 = a2 * s_w[778 + kk2];
    float* s_ctx = s_buf + 1536;
    s_ctx[kk0 * C_ + c] = ctx0;
    s_ctx[kk1 * C_ + c] = ctx1;
    s_ctx[kk2 * C_ + c] = ctx2;
    __syncthreads();
    if (t < 192) {
        int k = t >> 5, ln = t & 31;
        float s1 = 0.f, s2 = 0.f;
        #pragma unroll
        for (int i = 0; i < 4; ++i) {
            float v = s_ctx[k * C_ + ln + 32 * i];
            s1 += v; s2 += v * v;
        }
        s_buf[t] = s1; s_buf[256 + t] = s2;
    }
    __syncthreads();
    if (t < K_) {
        float s1 = 0.f, s2 = 0.f;
        for (int ln = 0; ln < 32; ++ln) { s1 += s_buf[t * 32 + ln]; s2 += s_buf[256 + t * 32 + ln]; }
        float mu = s1 * (1.0f / C_);
        float var = s2 * (1.0f / C_) - mu * mu;
        s_buf[512 + t * 2]     = mu;
        s_buf[512 + t * 2 + 1] = rsqrtf(var + EPS_);
    }
    __syncthreads();
    float g = ln1_g[c], bb = ln1_b[c];
    float x0 = (ctx0 - s_buf[512 + kk0 * 2]) * s_buf[512 + kk0 * 2 + 1] * g + bb;
    float x1 = (ctx1 - s_buf[512 + kk1 * 2]) * s_buf[512 + kk1 * 2 + 1] * g + bb;
    float x2 = (ctx2 - s_buf[512 + kk2 * 2]) * s_buf[512 + kk2 * 2 + 1] * g + bb;
    center[((size_t)kk0 * NPATCH + bp) * C_ + c] = gelu_exact(x0);
    center[((size_t)kk1 * NPATCH + bp) * C_ + c] = gelu_exact(x1);
    center[((size_t)kk2 * NPATCH + bp) * C_ + c] = gelu_exact(x2);
}

// =====================================================================
// K2: per class k: nearest-prototype assignment + scatter sum + EMA
// 6 blocks x 256 threads (thread = patch)
// =====================================================================
__global__ void k2_proto_update(const float* __restrict__ center,
                                const float* __restrict__ prototype,
                                float* __restrict__ proto_upd) {
    __shared__ float s_p[P_ * C_];   // prototypes for this k
    __shared__ float s_a[P_ * C_];   // scatter accumulators
    const int k = blockIdx.x;
    const int t = threadIdx.x;
    for (int i = t; i < P_ * C_; i += 256) {
        int p = i >> 7, c = i & 127;
        s_p[i] = prototype[((size_t)p * K_ + k) * C_ + c];
        s_a[i] = 0.f;
    }
    __syncthreads();

    const float* cen = center + ((size_t)k * NPATCH + t) * C_;
    float d2[P_] = {0, 0, 0, 0, 0, 0, 0, 0};
    for (int c = 0; c < C_; ++c) {
        float f = cen[c];
        #pragma unroll
        for (int p = 0; p < P_; ++p) {
            float df = f - s_p[p * C_ + c];
            d2[p] = fmaf(df, df, d2[p]);
        }
    }
    int pmin = 0; float dmin = d2[0];
    #pragma unroll
    for (int p = 1; p < P_; ++p) if (d2[p] < dmin) { dmin = d2[p]; pmin = p; }
    for (int c = 0; c < C_; ++c) atomicAdd(&s_a[pmin * C_ + c], cen[c]);   // ds_add_f32
    __syncthreads();

    // proto_upd[p*K+k] = M*proto + (1-M)*new_proto
    for (int i = t; i < P_ * C_; i += 256) {
        int p = i >> 7, c = i & 127;
        size_t row = (size_t)p * K_ + k;
        proto_upd[row * C_ + c] = 0.999f * prototype[row * C_ + c] + 0.001f * s_a[i];
    }
}

// =====================================================================
// K3: WMMA fp32 distance GEMM (M=48, K=128, N=B*HW) fused with
//     sim = 1/(1+2d) -> LN(48) -> GELU -> max over P  -> out_small
// grid = B * (HW/128) = 2048 blocks x 256 threads; wave owns 48x16 tile
// =====================================================================
__global__ void k3_sim_wmma(const float* __restrict__ feats,
                            const float* __restrict__ proto_upd,
                            const float* __restrict__ ln2_g,
                            const float* __restrict__ ln2_b,
                            float* __restrict__ out_small) {
    __shared__ float s_A[KP_ * 130];        // protos, rows padded to 130 (bank-safe)
    __shared__ float s_n2[KP_];             // |proto_j|^2
    __shared__ float s_sim[8 * KP_ * 16];   // per-wave 48x16 sim tiles

    const int t = threadIdx.x;
    for (int i = t; i < KP_ * C_; i += 256)
        s_A[(i >> 7) * 130 + (i & 127)] = proto_upd[i];
    __syncthreads();
    if (t < KP_) {
        float s = 0.f;
        for (int c = 0; c < C_; ++c) { float v = s_A[t * 130 + c]; s = fmaf(v, v, s); }
        s_n2[t] = s;
    }
    __syncthreads();

    const int b       = blockIdx.x >> 7;
    const int pixbase = (blockIdx.x & 127) * 128;
    const int wave = t >> 5, lane = t & 31;
    const int half = lane >> 4, pcol = lane & 15;
    const int pix  = pixbase + wave * 16 + pcol;
    const float* fb = feats + (size_t)b * C_ * HW_ + pix;

    v8f acc0 = {0,0,0,0,0,0,0,0};
    v8f acc1 = {0,0,0,0,0,0,0,0};
    v8f acc2 = {0,0,0,0,0,0,0,0};
    float bsq = 0.f;
    const float* arow = s_A + pcol * 130 + 2 * half;   // A frag base for M-tile 0

    for (int kk = 0; kk < 32; ++kk) {                  // K stepped by 4
        int c0 = kk * 4 + 2 * half;
        float bb0 = fb[(size_t)c0 * HW_];
        float bb1 = fb[(size_t)(c0 + 1) * HW_];
        bsq = fmaf(bb0, bb0, fmaf(bb1, bb1, bsq));
        v2f bf; bf.x = bb0; bf.y = bb1;
        const float* ap = arow + kk * 4;
        v2f a0; a0.x = ap[0];          a0.y = ap[1];
        v2f a1; a1.x = ap[16 * 130];   a1.y = ap[16 * 130 + 1];
        v2f a2; a2.x = ap[32 * 130];   a2.y = ap[32 * 130 + 1];
        acc0 = __builtin_amdgcn_wmma_f32_16x16x4_f32(false, a0, false, bf, (short)0, acc0, false, false);
        acc1 = __builtin_amdgcn_wmma_f32_16x16x4_f32(false, a1, false, bf, (short)0, acc1, false, false);
        acc2 = __builtin_amdgcn_wmma_f32_16x16x4_f32(false, a2, false, bf, (short)0, acc2, false, false);
    }
    // full |pixel|^2: lane halves each saw K%4 in {0,1} or {2,3}
    float bsqf = bsq + __shfl_xor(bsq, 16, 32);

    float* sw = s_sim + wave * KP_ * 16;
    #pragma unroll
    for (int v = 0; v < 8; ++v) {
        {
            int j = v + 8 * half;
            float d2 = s_n2[j] + bsqf - 2.0f * acc0[v];
            float d = sqrtf(fmaxf(d2, 0.f));
            sw[j * 16 + pcol] = 1.0f / (1.0f + 2.0f * d);
        }
        {
            int j = 16 + v + 8 * half;
            float d2 = s_n2[j] + bsqf - 2.0f * acc1[v];
            float d = sqrtf(fmaxf(d2, 0.f));
            sw[j * 16 + pcol] = 1.0f / (1.0f + 2.0f * d);
        }
        {
            int j = 32 + v + 8 * half;
            float d2 = s_n2[j] + bsqf - 2.0f * acc2[v];
            float d = sqrtf(fmaxf(d2, 0.f));
            sw[j * 16 + pcol] = 1.0f / (1.0f + 2.0f * d);
        }
    }
    __syncthreads();

    if (half == 0) {                        // 16 lanes = 16 pixels per wave
        float vals[KP_];
        float s1 = 0.f, s2 = 0.f;
        #pragma unroll
        for (int j = 0; j < KP_; ++j) {
            float s = sw[j * 16 + pcol];
            vals[j] = s; s1 += s; s2 += s * s;
        }
        float mu = s1 * (1.0f / KP_);
        float rs = rsqrtf(s2 * (1.0f / KP_) - mu * mu + EPS_);
        float mx[K_];
        #pragma unroll
        for (int k = 0; k < K_; ++k) mx[k] = -3.4e38f;
        #pragma unroll
        for (int j = 0; j < KP_; ++j) {      // j = p*K + k
            float x = (vals[j] - mu) * rs * ln2_g[j] + ln2_b[j];
            float gl = gelu_exact(x);
            int k = j % K_;
            mx[k] = fmaxf(mx[k], gl);
        }
        size_t ob = (size_t)b * K_ * HW_ + pix;
        #pragma unroll
        for (int k = 0; k < K_; ++k) out_small[ob + (size_t)k * HW_] = mx[k];
    }
}

// =====================================================================
// K4: bilinear resize (B,6,128,128) -> (B,6,512,512), half-pixel + clamp
// =====================================================================
__global__ void k4_resize(const float* __restrict__ src, float* __restrict__ dst,
                          long long total) {
    long long idx = (long long)blockIdx.x * 256 + threadIdx.x;
    if (idx >= total) return;
    int ox = (int)(idx & 511);
    int oy = (int)((idx >> 9) & 511);
    int ch = (int)(idx >> 18);             // b*6 + k, 0..95
    float sx = (ox + 0.5f) * 0.25f - 0.5f;
    float sy = (oy + 0.5f) * 0.25f - 0.5f;
    float fx = floorf(sx), fy = floorf(sy);
    float wx = sx - fx, wy = sy - fy;
    int x0 = (int)fx, y0 = (int)fy;
    int x1 = min(x0 + 1, W_ - 1), y1 = min(y0 + 1, H_ - 1);
    x0 = max(x0, 0); y0 = max(y0, 0);
    const float* s = src + (size_t)ch * HW_;
    float v00 = s[y0 * W_ + x0], v01 = s[y0 * W_ + x1];
    float v10 = s[y1 * W_ + x0], v11 = s[y1 * W_ + x1];
    float v0 = v00 + wx * (v01 - v00);
    float v1 = v10 + wx * (v11 - v10);
    dst[idx] = v0 + wy * (v1 - v0);
}

// =====================================================================
extern "C" void kernel_launch(void* const* d_in, const int* in_sizes, int n_in,
                              void* d_out, int out_size, void* d_ws, size_t ws_size,
                              hipStream_t stream) {
    const float* feats     = (const float*)d_in[0];
    const float* coarse_w  = (const float*)d_in[1];
    const float* coarse_b  = (const float*)d_in[2];
    const float* ln1_g     = (const float*)d_in[3];
    const float* ln1_b     = (const float*)d_in[4];
    const float* ln2_g     = (const float*)d_in[5];
    const float* ln2_b     = (const float*)d_in[6];
    const float* prototype = (const float*)d_in[7];
    float* out = (float*)d_out;

    // workspace carve (floats): center(6*256*128) | proto_upd(48*128) | out_small(16*6*128*128)
    float* ws_center = (float*)d_ws;
    float* ws_proto  = ws_center + (size_t)K_ * NPATCH * C_;      // +196608
    float* ws_small  = ws_proto  + (size_t)KP_ * C_;              // +6144
    (void)in_sizes; (void)n_in; (void)ws_size; (void)out_size;

    k1_patch_ctx<<<NPATCH, 256, 0, stream>>>(feats, coarse_w, coarse_b, ln1_g, ln1_b, ws_center);
    k2_proto_update<<<K_, 256, 0, stream>>>(ws_center, prototype, ws_proto);
    k3_sim_wmma<<<B_ * (HW_ / 128), 256, 0, stream>>>(feats, ws_proto, ln2_g, ln2_b, ws_small);
    long long total = (long long)B_ * K_ * 512 * 512;             // 25,165,824
    k4_resize<<<(unsigned)((total + 255) / 256), 256, 0, stream>>>(ws_small, out, total);
}